// VQVAE_89146341196333
// MI455X (gfx1250) — compile-verified
//
#include <hip/hip_runtime.h>

typedef float v2f __attribute__((ext_vector_type(2)));
typedef float v8f __attribute__((ext_vector_type(8)));

__device__ __forceinline__ float silu_f(float x) { return x / (1.0f + __expf(-x)); }

// ---------------------------------------------------------------------------
// Generic implicit-GEMM conv1d via fp32 WMMA (16x16x4).
//  y[b, co, n] = bias[co] + sum_{ci,kt} w[co,ci,kt] * xd[b, ci, n*stride + kt - pad]
//  where xd is the (optionally ldil-dilated) input: xd[p] = x[p/ldil] if p%ldil==0 else 0.
//  Block = 128 threads (4 waves); each wave owns a 16-row Cout tile; the 16-col
//  input window (dilated domain) is staged once into LDS and shared by all waves.
// Requirements (all satisfied by this network): Lout%16==0, Cout%64==0 or ==64,
//  (Cin*K)%4==0, Cin<=256, window<=36.
// ---------------------------------------------------------------------------
#define CONV_MAX_LDSF (256 * 36)

__global__ __launch_bounds__(128)
void vqvae_conv_wmma(const float* __restrict__ x, const float* __restrict__ w,
                     const float* __restrict__ bias, float* __restrict__ y,
                     int Cin, int Lin, int Cout, int Lout,
                     int K, int stride, int pad, int ldil, int fuse_silu)
{
    __shared__ float sX[CONV_MAX_LDSF];
    const int tid    = threadIdx.x;
    const int lane   = tid & 31;
    const int wave   = tid >> 5;
    const int lanelo = lane & 15;
    const int hi     = lane >> 4;

    const int n0      = blockIdx.x * 16;
    const int b       = blockIdx.y;
    const int rowBase = blockIdx.z * 64 + wave * 16;

    const int Wd = 15 * stride + K;          // input window width (dilated domain)
    const int p0 = n0 * stride - pad;
    const int Ld = (Lin - 1) * ldil + 1;

    const float* xb = x + (size_t)b * Cin * Lin;

    // ---- stage input window into LDS (zeros for OOB / dilation holes) ----
    const int total = Cin * Wd;
    for (int idx = tid; idx < total; idx += 128) {
        int ci  = idx / Wd;
        int off = idx - ci * Wd;
        int p   = p0 + off;
        float v = 0.0f;
        if (p >= 0 && p < Ld) {
            if (ldil == 1)          v = xb[(size_t)ci * Lin + p];
            else if ((p & 1) == 0)  v = xb[(size_t)ci * Lin + (p >> 1)];
        }
        sX[ci * Wd + off] = v;
    }
    __syncthreads();

    const int KK     = Cin * K;
    const int m      = rowBase + lanelo;           // A-fragment row for this lane
    const int kOff   = hi ? 2 : 0;                 // K-slots {kOff, kOff+1}
    const int coloff = lanelo * stride;            // this lane's column offset (dilated)

    // incremental (ci, kt) for kp = kk + kOff
    int ci = kOff / K;
    int kt = kOff - ci * K;

    const float* wrow = w + (size_t)m * KK;

    v8f acc = {};
    for (int kk = 0; kk < KK; kk += 4) {
        const int kp = kk + kOff;
        v2f a;
        a.x = wrow[kp];
        a.y = wrow[kp + 1];
        int kt1 = kt + 1, ci1 = ci;
        if (kt1 == K) { kt1 = 0; ci1 = ci + 1; }
        v2f bf;
        bf.x = sX[ci  * Wd + coloff + kt ];
        bf.y = sX[ci1 * Wd + coloff + kt1];
        acc = __builtin_amdgcn_wmma_f32_16x16x4_f32(false, a, false, bf,
                                                    (short)0, acc, false, false);
        kt += 4;
        while (kt >= K) { kt -= K; ci += 1; }
    }

    const int ncol = n0 + lanelo;
    #pragma unroll
    for (int v = 0; v < 8; ++v) {
        const int mr = rowBase + v + (hi ? 8 : 0);     // C layout: VGPR v -> M = v + 8*hi
        float r = acc[v] + bias[mr];
        if (fuse_silu) r = silu_f(r);
        y[((size_t)b * Cout + mr) * Lout + ncol] = r;
    }
}

// ---------------------------------------------------------------------------
// fe: conv1d(Cin=1 -> C, k3, pad1) + SiLU  (K dim too small for WMMA)
// ---------------------------------------------------------------------------
__global__ void vqvae_conv_fe(const float* __restrict__ x, const float* __restrict__ w,
                              const float* __restrict__ bias, float* __restrict__ y,
                              int T_, int C)
{
    size_t i = (size_t)blockIdx.x * blockDim.x + threadIdx.x;
    size_t total = (size_t)32 * C * T_;
    if (i >= total) return;
    int l = (int)(i % T_);
    int c = (int)((i / T_) % C);
    int b = (int)(i / ((size_t)C * T_));
    const float* xb = x + (size_t)b * T_;
    float s = bias[c];
    #pragma unroll
    for (int k = 0; k < 3; ++k) {
        int p = l + k - 1;
        if (p >= 0 && p < T_) s += w[c * 3 + k] * xb[p];
    }
    y[i] = silu_f(s);
}

// ---------------------------------------------------------------------------
// fr: convT1d(C -> 1, k3, pad1): out[b,l] = b0 + sum_ci sum_k w[ci,0,2-k]*x[b,ci,l+k-1]
// ---------------------------------------------------------------------------
__global__ void vqvae_conv_fr(const float* __restrict__ x, const float* __restrict__ w,
                              const float* __restrict__ bias, float* __restrict__ y,
                              int C, int T_)
{
    size_t i = (size_t)blockIdx.x * blockDim.x + threadIdx.x;
    if (i >= (size_t)32 * T_) return;
    int l = (int)(i % T_);
    int b = (int)(i / T_);
    const float* xb = x + (size_t)b * C * T_;
    float s = bias[0];
    for (int ci = 0; ci < C; ++ci) {
        const float* xc = xb + (size_t)ci * T_;
        #pragma unroll
        for (int k = 0; k < 3; ++k) {
            int p = l + k - 1;
            if (p >= 0 && p < T_) s += w[ci * 3 + (2 - k)] * xc[p];
        }
    }
    y[i] = s;
}

// ---------------------------------------------------------------------------
// ConvTranspose weight repack: wr[co][ci][k] = w[ci][co][K-1-k]
// ---------------------------------------------------------------------------
__global__ void vqvae_repack(const float* __restrict__ w, float* __restrict__ wr,
                             int Cin, int Cout, int K)
{
    int i = blockIdx.x * blockDim.x + threadIdx.x;
    int total = Cin * Cout * K;
    if (i < total) {
        int k  = i % K;
        int co = (i / K) % Cout;
        int ci = i / (K * Cout);
        wr[((size_t)co * Cin + ci) * K + k] = w[((size_t)ci * Cout + co) * K + (K - 1 - k)];
    }
}

// ---------------------------------------------------------------------------
// BatchNorm: stats (one block per channel, double accumulation) + apply
// Optional residual addend h for "batchnorm(out + h)".
// ---------------------------------------------------------------------------
__global__ void vqvae_bn_stats(const float* __restrict__ xx, const float* __restrict__ hh,
                               float* __restrict__ mean, float* __restrict__ rstd,
                               int C, int L)
{
    __shared__ double s1[256], s2[256];
    const int c = blockIdx.x;
    double a = 0.0, q = 0.0;
    const int n = 32 * L;
    for (int i = threadIdx.x; i < n; i += 256) {
        int b = i / L, l = i - b * L;
        size_t idx = ((size_t)b * C + c) * L + l;
        float v = xx[idx];
        if (hh) v += hh[idx];
        a += (double)v;
        q += (double)v * (double)v;
    }
    s1[threadIdx.x] = a; s2[threadIdx.x] = q;
    __syncthreads();
    for (int st = 128; st > 0; st >>= 1) {
        if (threadIdx.x < st) { s1[threadIdx.x] += s1[threadIdx.x + st];
                                s2[threadIdx.x] += s2[threadIdx.x + st]; }
        __syncthreads();
    }
    if (threadIdx.x == 0) {
        double m   = s1[0] / n;
        double var = s2[0] / n - m * m;
        mean[c] = (float)m;
        rstd[c] = rsqrtf((float)var + 1e-5f);
    }
}

__global__ void vqvae_bn_apply(const float* __restrict__ xx, const float* __restrict__ hh,
                               const float* __restrict__ mean, const float* __restrict__ rstd,
                               const float* __restrict__ g, const float* __restrict__ bb,
                               float* __restrict__ y, int C, int L, size_t total, int fuse_silu)
{
    size_t i = (size_t)blockIdx.x * blockDim.x + threadIdx.x;
    if (i >= total) return;
    int c = (int)((i / (size_t)L) % C);
    float v = xx[i];
    if (hh) v += hh[i];
    v = (v - mean[c]) * rstd[c] * g[c] + bb[c];
    if (fuse_silu) v = silu_f(v);
    y[i] = v;
}

// ---------------------------------------------------------------------------
// (B,C,L) -> (B*L, C) tiled transpose
// ---------------------------------------------------------------------------
__global__ void vqvae_transpose(const float* __restrict__ zin, float* __restrict__ zT,
                                int C, int L)
{
    __shared__ float tile[32][33];
    const int b  = blockIdx.z;
    const int l0 = blockIdx.x * 32;
    const int c0 = blockIdx.y * 32;
    const int lx = threadIdx.x, ly = threadIdx.y;
    for (int i = ly; i < 32; i += 8)
        tile[i][lx] = zin[((size_t)b * C + c0 + i) * L + l0 + lx];
    __syncthreads();
    for (int i = ly; i < 32; i += 8)
        zT[((size_t)b * L + l0 + i) * C + c0 + lx] = tile[lx][i];
}

// ---------------------------------------------------------------------------
// VQ: codebook column norms
// ---------------------------------------------------------------------------
__global__ void vqvae_embed_norm(const float* __restrict__ embed, float* __restrict__ enorm)
{
    int j = blockIdx.x * blockDim.x + threadIdx.x;
    if (j < 512) {
        float s = 0.0f;
        for (int c = 0; c < 128; ++c) { float e = embed[c * 512 + j]; s += e * e; }
        enorm[j] = s;
    }
}

// ---------------------------------------------------------------------------
// VQ argmin: scores = zT @ embed via WMMA; dist = ||e||^2 - 2*score.
// Block = 256 thr (8 waves), one 16-row M tile; wave w covers col tiles w,w+8,w+16,w+24.
// ---------------------------------------------------------------------------
__global__ __launch_bounds__(256)
void vqvae_vq_argmin(const float* __restrict__ zT, const float* __restrict__ embed,
                     const float* __restrict__ enorm, int* __restrict__ ind)
{
    const int tid = threadIdx.x;
    const int lane = tid & 31, wave = tid >> 5;
    const int lanelo = lane & 15, hi = lane >> 4;
    const int m0 = blockIdx.x * 16;
    const int kOff = hi ? 2 : 0;

    const float* zrow = zT + (size_t)(m0 + lanelo) * 128;

    float bestv[8]; int besti[8];
    #pragma unroll
    for (int v = 0; v < 8; ++v) { bestv[v] = 3.4e38f; besti[v] = 0; }

    for (int t = 0; t < 4; ++t) {
        const int nt = wave + t * 8;
        const int ncol = nt * 16 + lanelo;
        v8f acc = {};
        for (int kk = 0; kk < 128; kk += 4) {
            v2f a, bf;
            a.x  = zrow[kk + kOff];
            a.y  = zrow[kk + kOff + 1];
            bf.x = embed[(kk + kOff    ) * 512 + ncol];
            bf.y = embed[(kk + kOff + 1) * 512 + ncol];
            acc = __builtin_amdgcn_wmma_f32_16x16x4_f32(false, a, false, bf,
                                                        (short)0, acc, false, false);
        }
        const float en = enorm[ncol];
        #pragma unroll
        for (int v = 0; v < 8; ++v) {
            float d = en - 2.0f * acc[v];
            if (d < bestv[v] || (d == bestv[v] && ncol < besti[v])) { bestv[v] = d; besti[v] = ncol; }
        }
    }
    // argmin across the 16 lanes of each half (columns); rows stay resident per lane
    #pragma unroll
    for (int sh = 1; sh < 16; sh <<= 1) {
        #pragma unroll
        for (int v = 0; v < 8; ++v) {
            float ov = __shfl_xor(bestv[v], sh, 32);
            int   oi = __shfl_xor(besti[v], sh, 32);
            if (ov < bestv[v] || (ov == bestv[v] && oi < besti[v])) { bestv[v] = ov; besti[v] = oi; }
        }
    }
    __shared__ float sMin[8][16];
    __shared__ int   sIdx[8][16];
    if (lanelo == 0) {
        #pragma unroll
        for (int v = 0; v < 8; ++v) {
            int row = v + hi * 8;
            sMin[wave][row] = bestv[v];
            sIdx[wave][row] = besti[v];
        }
    }
    __syncthreads();
    if (tid < 16) {
        float bv = sMin[0][tid]; int bi = sIdx[0][tid];
        for (int wv = 1; wv < 8; ++wv) {
            float ov = sMin[wv][tid]; int oi = sIdx[wv][tid];
            if (ov < bv || (ov == bv && oi < bi)) { bv = ov; bi = oi; }
        }
        ind[m0 + tid] = bi;
    }
}

// q gather directly into (B, C, Lq) layout for the decoder
__global__ void vqvae_q_gather(const float* __restrict__ embed, const int* __restrict__ ind,
                               float* __restrict__ qT, int C, int L)
{
    size_t i = (size_t)blockIdx.x * blockDim.x + threadIdx.x;
    size_t total = (size_t)32 * C * L;
    if (i >= total) return;
    int l = (int)(i % L);
    int c = (int)((i / L) % C);
    int b = (int)(i / ((size_t)C * L));
    qT[i] = embed[c * 512 + ind[b * L + l]];
}

// deterministic two-stage mean((a-b)^2)
__global__ void vqvae_sqdiff_part(const float* __restrict__ a, const float* __restrict__ b,
                                  float* __restrict__ part, size_t total)
{
    __shared__ double red[256];
    double s = 0.0;
    for (size_t i = (size_t)blockIdx.x * blockDim.x + threadIdx.x; i < total;
         i += (size_t)gridDim.x * blockDim.x) {
        float d = a[i] - b[i];
        s += (double)d * (double)d;
    }
    red[threadIdx.x] = s;
    __syncthreads();
    for (int st = 128; st > 0; st >>= 1) {
        if (threadIdx.x < st) red[threadIdx.x] += red[threadIdx.x + st];
        __syncthreads();
    }
    if (threadIdx.x == 0) part[blockIdx.x] = (float)red[0];
}

__global__ void vqvae_reduce_mean(const float* __restrict__ part, int n, float scale,
                                  float* __restrict__ out)
{
    __shared__ double red[256];
    double s = 0.0;
    for (int i = threadIdx.x; i < n; i += 256) s += (double)part[i];
    red[threadIdx.x] = s;
    __syncthreads();
    for (int st = 128; st > 0; st >>= 1) {
        if (threadIdx.x < st) red[threadIdx.x] += red[threadIdx.x + st];
        __syncthreads();
    }
    if (threadIdx.x == 0) *out = (float)(red[0] * (double)scale);
}

// ---------------------------------------------------------------------------
// Host orchestration
// ---------------------------------------------------------------------------
static inline void launch_conv(hipStream_t s, const float* x, const float* w, const float* b,
                               float* y, int Cin, int Lin, int Cout, int Lout,
                               int K, int stride, int pad, int ldil, int silu)
{
    dim3 grid(Lout / 16, 32, Cout / 64);
    vqvae_conv_wmma<<<grid, 128, 0, s>>>(x, w, b, y, Cin, Lin, Cout, Lout, K, stride, pad, ldil, silu);
}

static inline void launch_bn(hipStream_t s, const float* x, const float* h,
                             const float* g, const float* bb, float* y,
                             float* mean, float* rstd, int C, int L, int silu)
{
    vqvae_bn_stats<<<C, 256, 0, s>>>(x, h, mean, rstd, C, L);
    size_t total = (size_t)32 * C * L;
    vqvae_bn_apply<<<(unsigned)((total + 255) / 256), 256, 0, s>>>(x, h, mean, rstd, g, bb, y,
                                                                   C, L, total, silu);
}

static inline void launch_repack(hipStream_t s, const float* w, float* wr, int Cin, int Cout, int K)
{
    int total = Cin * Cout * K;
    vqvae_repack<<<(total + 255) / 256, 256, 0, s>>>(w, wr, Cin, Cout, K);
}

extern "C" void kernel_launch(void* const* d_in, const int* in_sizes, int n_in,
                              void* d_out, int out_size, void* d_ws, size_t ws_size,
                              hipStream_t stream)
{
    (void)in_sizes; (void)n_in; (void)out_size; (void)ws_size;
    int ix = 0;
    auto in = [&](void) { return (const float*)d_in[ix++]; };

    const float* X   = in();                 // (32,1,4096)
    const float* feW = in(); const float* feB = in();
    const float *eaW1[2], *eaB1[2], *eaW2[2], *eaB2[2], *eaW3[2], *eaB3[2], *eaG[2], *eaBB[2];
    for (int i = 0; i < 2; ++i) { eaW1[i]=in(); eaB1[i]=in(); eaW2[i]=in(); eaB2[i]=in();
                                  eaW3[i]=in(); eaB3[i]=in(); eaG[i]=in(); eaBB[i]=in(); }
    const float *erW1[2], *erB1[2], *erW2[2], *erB2[2], *erG[2], *erBB[2];
    for (int i = 0; i < 2; ++i) { erW1[i]=in(); erB1[i]=in(); erW2[i]=in(); erB2[i]=in();
                                  erG[i]=in(); erBB[i]=in(); }
    const float* fiW = in(); const float* fiB = in();
    const float* EMB = in();                 // (128,512)
    const float* dfW = in(); const float* dfB = in();
    const float *drW1[2], *drB1[2], *drW2[2], *drB2[2], *drG[2], *drBB[2];
    for (int i = 0; i < 2; ++i) { drW1[i]=in(); drB1[i]=in(); drW2[i]=in(); drB2[i]=in();
                                  drG[i]=in(); drBB[i]=in(); }
    const float *daW1[2], *daB1[2], *daW2[2], *daB2[2], *daW3[2], *daB3[2], *daG[2], *daBB[2];
    for (int i = 0; i < 2; ++i) { daW1[i]=in(); daB1[i]=in(); daW2[i]=in(); daB2[i]=in();
                                  daW3[i]=in(); daB3[i]=in(); daG[i]=in(); daBB[i]=in(); }
    const float* frW = in(); const float* frB = in();

    // workspace layout (floats)
    const size_t QE = 16777216ull;           // 32*128*4096 == 32*256*2048
    float* W  = (float*)d_ws;
    float* A0 = W;                           // QE
    float* A1 = W + QE;                      // QE
    float* BB = W + 2 * QE;                  // 2*QE (also aliases ZT during VQ)
    float* WR    = W + 4 * QE;               // 262144 (repack scratch)
    float* MEAN  = WR + 262144;              // 256
    float* RSTD  = MEAN + 256;               // 256
    float* ENORM = RSTD + 256;               // 512
    float* PART  = ENORM + 512;              // 1024
    int*   IND   = (int*)(PART + 1024);      // 32768
    float* ZT    = BB;
    float* OUT   = (float*)d_out;

    // ================= Encoder =================
    {
        size_t total = (size_t)32 * 128 * 4096;
        vqvae_conv_fe<<<(unsigned)((total + 255) / 256), 256, 0, stream>>>(X, feW, feB, A0, 4096, 128);
    }
    int L = 4096;
    float* cur = A0;
    for (int blk = 0; blk < 2; ++blk) {
        float* other = (cur == A0) ? A1 : A0;
        launch_conv(stream, cur,   eaW1[blk], eaB1[blk], BB,    128, L,     256, L,     3, 1, 1, 1, 1);
        launch_conv(stream, BB,    eaW2[blk], eaB2[blk], cur,   256, L,     256, L / 2, 4, 2, 1, 1, 1);
        L /= 2;
        launch_conv(stream, cur,   eaW3[blk], eaB3[blk], other, 256, L,     128, L,     3, 1, 1, 1, 0);
        launch_bn(stream, other, nullptr, eaG[blk], eaBB[blk], other, MEAN, RSTD, 128, L, 1);
        cur = other;                        // blk0 -> A1, blk1 -> A0; L = 1024
    }
    for (int r = 0; r < 2; ++r) {
        float* other = (cur == A0) ? A1 : A0;
        launch_conv(stream, cur,   erW1[r], erB1[r], other, 128, L, 64,  L, 3, 1, 1, 1, 1);
        launch_conv(stream, other, erW2[r], erB2[r], BB,    64,  L, 128, L, 1, 1, 0, 1, 1);
        launch_bn(stream, cur, BB, erG[r], erBB[r], other, MEAN, RSTD, 128, L, 0);
        cur = other;                        // r0 -> A1, r1 -> A0
    }
    launch_conv(stream, cur, fiW, fiB, A1, 128, L, 128, L, 3, 1, 1, 1, 0);   // z in A1

    // ================= Quantize =================
    {
        dim3 grid(L / 32, 128 / 32, 32);
        vqvae_transpose<<<grid, dim3(32, 8), 0, stream>>>(A1, ZT, 128, L);
    }
    vqvae_embed_norm<<<2, 256, 0, stream>>>(EMB, ENORM);
    vqvae_vq_argmin<<<(32 * L) / 16, 256, 0, stream>>>(ZT, EMB, ENORM, IND);
    {
        size_t total = (size_t)32 * 128 * L;
        vqvae_q_gather<<<(unsigned)((total + 255) / 256), 256, 0, stream>>>(EMB, IND, A0, 128, L);
        vqvae_sqdiff_part<<<1024, 256, 0, stream>>>(A1, A0, PART, total);
        vqvae_reduce_mean<<<1, 256, 0, stream>>>(PART, 1024, 1.0f / (float)total, OUT + 131072);
    }

    // ================= Decoder ================= (input q in A0)
    launch_conv(stream, A0, dfW, dfB, BB, 128, L, 128, L, 3, 1, 1, 1, 0);
    cur = BB;
    for (int r = 0; r < 2; ++r) {
        float* h1  = (r == 0) ? A0 : A1;     // 64-ch temp
        float* h2  = (r == 0) ? A1 : BB;     // 128-ch temp
        float* dst = (r == 0) ? A0 : A1;
        launch_repack(stream, drW1[r], WR, 128, 64, 3);
        launch_conv(stream, cur, WR, drB1[r], h1, 128, L, 64,  L, 3, 1, 1, 1, 1);
        launch_repack(stream, drW2[r], WR, 64, 128, 1);
        launch_conv(stream, h1,  WR, drB2[r], h2, 64,  L, 128, L, 1, 1, 0, 1, 1);
        launch_bn(stream, cur, h2, drG[r], drBB[r], dst, MEAN, RSTD, 128, L, 0);
        cur = dst;                          // r0 -> A0, r1 -> A1
    }
    for (int blk = 0; blk < 2; ++blk) {
        float* other = (cur == A0) ? A1 : A0;
        float* up    = (blk == 0) ? cur : BB;
        float* dst   = (blk == 0) ? other : cur;
        launch_repack(stream, daW1[blk], WR, 128, 256, 3);
        launch_conv(stream, cur,   WR, daB1[blk], other, 128, L, 256, L,     3, 1, 1, 1, 1);
        launch_repack(stream, daW2[blk], WR, 256, 256, 4);
        launch_conv(stream, other, WR, daB2[blk], up,    256, L, 256, 2 * L, 4, 1, 2, 2, 1);
        L *= 2;
        launch_repack(stream, daW3[blk], WR, 256, 128, 3);
        launch_conv(stream, up,    WR, daB3[blk], dst,   256, L, 128, L,     3, 1, 1, 1, 0);
        launch_bn(stream, dst, nullptr, daG[blk], daBB[blk], dst, MEAN, RSTD, 128, L, 1);
        cur = dst;                          // both blocks end at A0; L: 1024->2048->4096
    }
    {
        size_t total = (size_t)32 * L;
        vqvae_conv_fr<<<(unsigned)((total + 255) / 256), 256, 0, stream>>>(cur, frW, frB, OUT, 128, L);
    }
}